// encoder_union_4956392259720
// MI455X (gfx1250) — compile-verified
//
#include <hip/hip_runtime.h>

typedef __attribute__((ext_vector_type(2))) float v2f;
typedef __attribute__((ext_vector_type(8))) float v8f;

#define H1c 2
#define H0c 2
#define Oc 32
#define D0c 64
#define OHc 4

// ---------------------------------------------------------------- utilities
__device__ __forceinline__ unsigned fenc(float f) {
  unsigned u = __float_as_uint(f);
  return ((int)u < 0) ? ~u : (u ^ 0x80000000u);   // order-preserving encode
}
__device__ __forceinline__ float fdec(unsigned u) {
  return __uint_as_float(((int)u < 0) ? (u ^ 0x80000000u) : ~u);
}

__global__ void zero_ws_kernel(uint4* __restrict__ p, long n16) {
  long i = (long)blockIdx.x * blockDim.x + threadIdx.x;
  if (i < n16) p[i] = make_uint4(0u, 0u, 0u, 0u);
}

// -------------------------------------------------- 1) z_dst via f32 WMMA
// z_dst[n,j,o] = sum_d hier0[n,d] * W_dst[j,o,d]
// One wave: 16 rows x head j x all 32 output cols (two 16x16 tiles, A reused).
__global__ void zdst_wmma_kernel(const float* __restrict__ hier0,
                                 const float* __restrict__ Wdst,
                                 float* __restrict__ zdst, int N) {
  int wave = (int)((blockIdx.x * (long)blockDim.x + threadIdx.x) >> 5);
  int lane = threadIdx.x & 31;
  int nTiles = (N + 15) >> 4;
  if (wave >= nTiles * H0c) return;          // wave-uniform exit
  int j    = wave % H0c;
  int row0 = (wave / H0c) << 4;

  // A fragment source: lanes 0-15 -> rows M=lane, K=4k+{0,1}; lanes 16-31 -> K=4k+{2,3}
  int m  = lane & 15;
  int kb = (lane >> 4) << 1;                 // 0 or 2
  int row = row0 + m; if (row >= N) row = N - 1;
  const float* arow = hier0 + (long)row * D0c;
  // B fragment source: column o = (lane&15) for tile0, 16+(lane&15) for tile1
  const float* w0 = Wdst + ((long)j * Oc + m) * D0c;
  const float* w1 = w0 + 16 * D0c;

  v8f acc0 = {}; v8f acc1 = {};
#pragma unroll
  for (int kk = 0; kk < D0c / 4; ++kk) {
    int k = kk * 4 + kb;
    v2f a, b0, b1;
    a.x  = arow[k]; a.y  = arow[k + 1];
    b0.x = w0[k];   b0.y = w0[k + 1];
    b1.x = w1[k];   b1.y = w1[k + 1];
    acc0 = __builtin_amdgcn_wmma_f32_16x16x4_f32(false, a, false, b0,
                                                 (short)0, acc0, false, false);
    acc1 = __builtin_amdgcn_wmma_f32_16x16x4_f32(false, a, false, b1,
                                                 (short)0, acc1, false, false);
  }
  // C/D layout: lanes 0-15: M=v, N=lane ; lanes 16-31: M=v+8, N=lane-16
  int ocol  = lane & 15;
  int mbase = (lane >> 4) << 3;
#pragma unroll
  for (int v = 0; v < 8; ++v) {
    int rm = row0 + mbase + v;
    if (rm < N) {
      float* drow = zdst + ((long)rm * H0c + j) * Oc;
      drow[ocol]      = acc0[v];
      drow[16 + ocol] = acc1[v];
    }
  }
}

// -------------------------------------------------- 2) attention scores
__global__ void scores_kernel(const float* __restrict__ hier1,
                              const float* __restrict__ zdst,
                              const float* __restrict__ Wattn,
                              float* __restrict__ ssrc, float* __restrict__ sdst,
                              int N) {
  int n = blockIdx.x * blockDim.x + threadIdx.x;
  if (n >= N) return;
  float acc[H1c][H0c] = {};
#pragma unroll
  for (int i = 0; i < H1c; ++i) {
    const float* z = hier1 + ((long)n * H1c + i) * Oc;
    for (int o = 0; o < Oc; ++o) {
      float zv = z[o];
#pragma unroll
      for (int j = 0; j < H0c; ++j) acc[i][j] += zv * Wattn[j * (2 * Oc) + o];
    }
  }
#pragma unroll
  for (int i = 0; i < H1c; ++i)
#pragma unroll
    for (int j = 0; j < H0c; ++j) ssrc[((long)n * H1c + i) * H0c + j] = acc[i][j];
#pragma unroll
  for (int j = 0; j < H0c; ++j) {
    const float* z = zdst + ((long)n * H0c + j) * Oc;
    float s = 0.f;
    for (int o = 0; o < Oc; ++o) s += z[o] * Wattn[j * (2 * Oc) + Oc + o];
    sdst[(long)n * H0c + j] = s;
  }
}

// -------------------------------------------------- 3) leaky-relu + segment max + deg
__global__ void edge_max_kernel(const int* __restrict__ src, const int* __restrict__ dst,
                                const float* __restrict__ ssrc, const float* __restrict__ sdst,
                                float* __restrict__ el, unsigned* __restrict__ menc,
                                float* __restrict__ deg, int E) {
  int e = blockIdx.x * blockDim.x + threadIdx.x;
  if (e >= E) return;
  int s = src[e], d = dst[e];
  float4 ss = *(const float4*)(ssrc + (long)s * 4);
  float sv[4] = {ss.x, ss.y, ss.z, ss.w};
  float sd0 = sdst[(long)d * 2], sd1 = sdst[(long)d * 2 + 1];
  float ev[4];
#pragma unroll
  for (int c = 0; c < 4; ++c) {
    float x = sv[c] + ((c & 1) ? sd1 : sd0);       // c = i*H0 + j ; j = c&1
    ev[c] = (x >= 0.f) ? x : 0.01f * x;            // leaky_relu slope 0.01
    atomicMax(menc + (long)d * 4 + c, fenc(ev[c]));
  }
  *(float4*)(el + (long)e * 4) = make_float4(ev[0], ev[1], ev[2], ev[3]);
  atomicAdd(deg + d, 1.0f);
}

// -------------------------------------------------- 4) exp + segment sum (denom)
__global__ void edge_exp_kernel(const int* __restrict__ dst, float* __restrict__ elex,
                                const unsigned* __restrict__ menc,
                                float* __restrict__ denom, int E) {
  int e = blockIdx.x * blockDim.x + threadIdx.x;
  if (e >= E) return;
  int d = dst[e];
  float4 v4 = *(const float4*)(elex + (long)e * 4);
  float ev[4] = {v4.x, v4.y, v4.z, v4.w};
#pragma unroll
  for (int c = 0; c < 4; ++c) {
    float mm = fdec(menc[(long)d * 4 + c]);
    float ex = __expf(ev[c] - mm);
    ev[c] = ex;
    atomicAdd(denom + (long)d * 4 + c, ex);
  }
  *(float4*)(elex + (long)e * 4) = make_float4(ev[0], ev[1], ev[2], ev[3]);
}

// -------------------------------------------------- 5) alpha * z_src scatter-add
// thread = (edge, o) ; 4 atomic adds per thread (i,j combos)
__global__ void edge_scatter_kernel(const int* __restrict__ src, const int* __restrict__ dst,
                                    const float* __restrict__ hier1,
                                    const float* __restrict__ ex, const float* __restrict__ denom,
                                    float* __restrict__ agg, int E) {
  long tid = (long)blockIdx.x * blockDim.x + threadIdx.x;
  int e = (int)(tid >> 5);
  int o = (int)(tid & 31);
  if (e >= E) return;
  int s = src[e], d = dst[e];
  float a[4];
#pragma unroll
  for (int c = 0; c < 4; ++c) a[c] = ex[(long)e * 4 + c] / denom[(long)d * 4 + c];
  float z0 = hier1[((long)s * H1c + 0) * Oc + o];
  float z1 = hier1[((long)s * H1c + 1) * Oc + o];
  float* ag = agg + (long)d * (H1c * H0c * Oc) + o;
  atomicAdd(ag + 0 * Oc, a[0] * z0);
  atomicAdd(ag + 1 * Oc, a[1] * z0);
  atomicAdd(ag + 2 * Oc, a[2] * z1);
  atomicAdd(ag + 3 * Oc, a[3] * z1);
}

// -------------------------------------------------- 6) finalize + wn mix
__global__ void finalize_kernel(const float* __restrict__ zdst, const float* __restrict__ agg,
                                const float* __restrict__ deg, const float* __restrict__ Wwn,
                                float* __restrict__ out, int N) {
  long tid = (long)blockIdx.x * blockDim.x + threadIdx.x;
  int n = (int)(tid >> 5);
  int o = (int)(tid & 31);
  if (n >= N) return;
  float* op = out + (long)n * OHc * Oc + o;
  float dg = deg[n];
  if (dg <= 0.f) {
#pragma unroll
    for (int m2 = 0; m2 < OHc; ++m2) op[m2 * Oc] = 0.f;
    return;
  }
  float inv = 1.f / dg;                         // deg>=1 here, so safe==deg
  float h[4];
#pragma unroll
  for (int c = 0; c < 4; ++c) {
    int j = c & 1;
    h[c] = zdst[((long)n * H0c + j) * Oc + o] * inv +
           agg[(long)n * (H1c * H0c * Oc) + c * Oc + o];
  }
  float wn[OHc][4];
  float colsum[4] = {0.f, 0.f, 0.f, 0.f};
#pragma unroll
  for (int m2 = 0; m2 < OHc; ++m2)
#pragma unroll
    for (int k = 0; k < 4; ++k) {
      float w = Wwn[m2 * 4 + k];
      w = (w > 0.f) ? w : 0.f;
      wn[m2][k] = w;
      colsum[k] += w;
    }
#pragma unroll
  for (int m2 = 0; m2 < OHc; ++m2) {
    float sacc = 0.f;
#pragma unroll
    for (int k = 0; k < 4; ++k) sacc += (wn[m2][k] / colsum[k]) * h[k];
    op[m2 * Oc] = sacc;
  }
}

// ================================================================ launcher
extern "C" void kernel_launch(void* const* d_in, const int* in_sizes, int n_in,
                              void* d_out, int out_size, void* d_ws, size_t ws_size,
                              hipStream_t stream) {
  const float* hier1 = (const float*)d_in[0];
  const float* hier0 = (const float*)d_in[1];
  const int*   src   = (const int*)d_in[2];
  const int*   dst   = (const int*)d_in[3];
  const float* Wdst  = (const float*)d_in[4];
  const float* Wattn = (const float*)d_in[5];
  const float* Wwn   = (const float*)d_in[6];
  float* out = (float*)d_out;

  const int N = in_sizes[1] / D0c;
  const int E = in_sizes[2];

  // workspace partition (256B aligned)
  size_t off = 0;
  auto take = [&](size_t bytes) { size_t o = off; off = (off + bytes + 255) & ~(size_t)255; return o; };
  char* ws = (char*)d_ws;
  size_t o_zdst  = take((size_t)N * H0c * Oc * 4);
  size_t o_ssrc  = take((size_t)N * 4 * 4);
  size_t o_sdst  = take((size_t)N * 2 * 4);
  size_t o_el    = take((size_t)E * 4 * 4);
  size_t o_menc  = take((size_t)N * 4 * 4);          // zero region starts here
  size_t o_denom = take((size_t)N * 4 * 4);
  size_t o_agg   = take((size_t)N * H1c * H0c * Oc * 4);
  size_t o_deg   = take((size_t)N * 4);
  size_t zero_begin = o_menc;
  size_t zero_end   = off;                           // 256-aligned

  float*    zdst  = (float*)(ws + o_zdst);
  float*    ssrc  = (float*)(ws + o_ssrc);
  float*    sdstp = (float*)(ws + o_sdst);
  float*    el    = (float*)(ws + o_el);
  unsigned* menc  = (unsigned*)(ws + o_menc);
  float*    denom = (float*)(ws + o_denom);
  float*    agg   = (float*)(ws + o_agg);
  float*    deg   = (float*)(ws + o_deg);

  const int T = 256;

  // 0) zero: menc (0u == fenc-lowest sentinel: below every encoded value), denom, agg, deg
  long n16 = (long)((zero_end - zero_begin) >> 4);
  zero_ws_kernel<<<(int)((n16 + T - 1) / T), T, 0, stream>>>((uint4*)(ws + zero_begin), n16);

  // 1) z_dst projection on the matrix pipe (f32 WMMA, exact)
  int waves = ((N + 15) >> 4) * H0c;
  zdst_wmma_kernel<<<(waves + 7) / 8, T, 0, stream>>>(hier0, Wdst, zdst, N);

  // 2) scores
  scores_kernel<<<(N + T - 1) / T, T, 0, stream>>>(hier1, zdst, Wattn, ssrc, sdstp, N);

  // 3) leaky-relu + segment max + degree
  edge_max_kernel<<<(E + T - 1) / T, T, 0, stream>>>(src, dst, ssrc, sdstp, el, menc, deg, E);

  // 4) exp + denom segment sum (overwrites el with ex)
  edge_exp_kernel<<<(E + T - 1) / T, T, 0, stream>>>(dst, el, menc, denom, E);

  // 5) softmax-weighted message scatter
  long sthreads = (long)E * 32;
  edge_scatter_kernel<<<(int)((sthreads + T - 1) / T), T, 0, stream>>>(src, dst, hier1, el, denom, agg, E);

  // 6) finalize + output mix
  long fthreads = (long)N * 32;
  finalize_kernel<<<(int)((fthreads + T - 1) / T), T, 0, stream>>>(zdst, agg, deg, Wwn, out, N);
}